// F_local_72335839200042
// MI455X (gfx1250) — compile-verified
//
#include <hip/hip_runtime.h>
#include <math.h>

typedef __attribute__((ext_vector_type(2))) float v2f;
typedef __attribute__((ext_vector_type(8))) float v8f;
typedef __attribute__((ext_vector_type(4))) unsigned int u32x4;
typedef __attribute__((ext_vector_type(8))) int i32x8;
typedef __attribute__((ext_vector_type(4))) int i32x4;

#define D_IN 992
#define H_   32
#define W1   6
#define NB   12              // G + K = 8 + 4 basis functions
#define ROWS_PER_BLOCK 128   // 8 waves * 16 rows
#define THREADS 256
#define HSTRIDE 33           // padded hidden row stride in LDS
#define KC 32                // K-chunk staged per TDM transfer
#define ASTRIDE 36           // KC + 4 dwords TDM padding -> conflict-free frags
#define NCHUNK (D_IN / KC)   // 31

#if __has_builtin(__builtin_amdgcn_tensor_load_to_lds)
#define HAVE_TDM 1
#else
#define HAVE_TDM 0
#endif

// ---------------------------------------------------------------------------
// TDM: issue a 2D tensor tile load (KC floats x ROWS_PER_BLOCK rows) from a
// row-major (.., D_IN) fp32 tensor into LDS with 4-dword padding per row
// (pad_interval code 4 = 32 dwords, pad_amount code 3 = 4 dwords).
// tensor_rows = rows remaining from tile start: TDM zero-fills OOB rows.
// ---------------------------------------------------------------------------
#if HAVE_TDM
__device__ __forceinline__ void tdm_load_A(const float* gsrc, unsigned lds_off,
                                           unsigned tensor_rows) {
    unsigned long long ga = (unsigned long long)(uintptr_t)gsrc;
    u32x4 g0;
    g0[0] = 1u;                                        // count=1, load, user mode
    g0[1] = lds_off;                                   // lds_addr (bytes)
    g0[2] = (unsigned)(ga & 0xFFFFFFFFu);              // global_addr[31:0]
    g0[3] = (unsigned)((ga >> 32) & 0x1FFFFFFu)        // global_addr[56:32]
          | (2u << 30);                                // type = 2 ("image")
    i32x8 g1;
    g1[0] = (int)((2u << 16)                           // data_size = 4 bytes
                | (1u << 20)                           // pad_enable
                | (4u << 22)                           // pad_interval: 32 dwords
                | (3u << 25));                         // pad_amount: 4 dwords
    g1[1] = (int)(((unsigned)D_IN & 0xFFFFu) << 16);   // tensor_dim0[15:0]
    g1[2] = (int)((((unsigned)D_IN >> 16) & 0xFFFFu)   // tensor_dim0[31:16]
                | ((tensor_rows & 0xFFFFu) << 16));    // tensor_dim1[15:0]
    g1[3] = (int)(((tensor_rows >> 16) & 0xFFFFu)      // tensor_dim1[31:16]
                | ((unsigned)KC << 16));               // tile_dim0 = KC
    g1[4] = (int)(ROWS_PER_BLOCK & 0xFFFF);            // tile_dim1 = 128, tile_dim2 = 0
    g1[5] = (int)D_IN;                                 // tensor_dim0_stride lo32
    g1[6] = 0;                                         // stride hi / dim1_stride
    g1[7] = 0;
    i32x4 z4 = {0, 0, 0, 0};
#if __clang_major__ >= 23
    i32x8 z8 = {0, 0, 0, 0, 0, 0, 0, 0};
    __builtin_amdgcn_tensor_load_to_lds(g0, g1, z4, z4, z8, 0);
#else
    __builtin_amdgcn_tensor_load_to_lds(g0, g1, z4, z4, 0);
#endif
}
#endif

// Degree-4 uniform B-spline basis on grid t_j = -2 + 0.25*j (17 points),
// Cox-de Boor, fully unrolled; constants fold at compile time.
__device__ __forceinline__ void bspline12(float x, float* Bv) {
    float w[16];
#pragma unroll
    for (int j = 0; j < 16; ++j) {
        float t0 = -2.0f + 0.25f * (float)j;
        float t1 = t0 + 0.25f;
        w[j] = (x >= t0 && x < t1) ? 1.0f : 0.0f;
    }
#pragma unroll
    for (int p = 1; p <= 4; ++p) {
        float inv = 4.0f / (float)p;   // 1 / (p*h), h = 0.25
#pragma unroll
        for (int j = 0; j < 16 - p; ++j) {
            float tj   = -2.0f + 0.25f * (float)j;      // t[j]
            float tj1p = tj + 0.25f * (float)(p + 1);   // t[j+p+1]
            w[j] = (x - tj) * inv * w[j] + (tj1p - x) * inv * w[j + 1];
        }
    }
#pragma unroll
    for (int b = 0; b < NB; ++b) Bv[b] = w[b];
}

__device__ __forceinline__ float silu(float x) {
    return x / (1.0f + __expf(-x));
}

__global__ __launch_bounds__(THREADS) void kan_fused_kernel(
    const float* __restrict__ node_rep,  // (N, 992)
    const float* __restrict__ mlp_w,     // (992, 32)
    const float* __restrict__ mlp_b,     // (32)
    const float* __restrict__ coef0,     // (32, 6, 12)
    const float* __restrict__ wb0,       // (32, 6)
    const float* __restrict__ ws0,       // (32, 6)
    const float* __restrict__ b0,        // (6)
    const float* __restrict__ coef1,     // (6, 1, 12)
    const float* __restrict__ wb1,       // (6, 1)
    const float* __restrict__ ws1,       // (6, 1)
    const float* __restrict__ b1,        // (1)
    float* __restrict__ out,             // (N, 1)
    int Nrows)
{
    __shared__ float sh_A[2][ROWS_PER_BLOCK * ASTRIDE];   // TDM-staged A chunks
    __shared__ float sh_hidden[ROWS_PER_BLOCK * HSTRIDE];
    __shared__ float sh_coef0[H_ * W1 * NB];
    __shared__ float sh_wb0[H_ * W1];
    __shared__ float sh_ws0[H_ * W1];
    __shared__ float sh_coef1[W1 * NB];
    __shared__ float sh_b0[W1];
    __shared__ float sh_wb1[W1];
    __shared__ float sh_ws1[W1];
    __shared__ float sh_mlpb[H_];
    __shared__ float sh_b1;

    const int tid  = threadIdx.x;
    const int wave = tid >> 5;
    const int lane = tid & 31;

    // ---- stage small parameters into LDS (covered by first barrier) ----
    for (int i = tid; i < H_ * W1 * NB; i += THREADS) sh_coef0[i] = coef0[i];
    for (int i = tid; i < H_ * W1; i += THREADS) { sh_wb0[i] = wb0[i]; sh_ws0[i] = ws0[i]; }
    for (int i = tid; i < W1 * NB; i += THREADS) sh_coef1[i] = coef1[i];
    if (tid < W1) { sh_b0[tid] = b0[tid]; sh_wb1[tid] = wb1[tid]; sh_ws1[tid] = ws1[tid]; }
    if (tid < H_) sh_mlpb[tid] = mlp_b[tid];
    if (tid == 0) sh_b1 = b1[0];

    const long long blockRowBase = (long long)blockIdx.x * ROWS_PER_BLOCK;
    const unsigned tensor_rows = (unsigned)(Nrows - blockRowBase);
    const float* gbase = node_rep + (size_t)blockRowBase * D_IN;

    // ---- phase 1: hidden = A @ mlp_w via V_WMMA_F32_16X16X4_F32 ----
    const int kq = (lane >> 4) * 2;               // lanes 0-15: K 0,1 ; 16-31: K 2,3
    // B 4x16 f32 layout: VGPR v, lane L -> K = kq + v, N = L%16 (+16 for tile 1)
    const float* bptr0 = mlp_w + (size_t)kq * H_ + (lane & 15);
    const float* bptr1 = bptr0 + 16;
    const int rowOff = (wave * 16 + (lane & 15)) * ASTRIDE + kq;

    v8f c0 = {};  // cols 0..15
    v8f c1 = {};  // cols 16..31

#if HAVE_TDM
    const bool issuer = (wave == 0);
    const unsigned ldsA0 = (unsigned)(uintptr_t)&sh_A[0][0];
    const unsigned ldsA1 = (unsigned)(uintptr_t)&sh_A[1][0];
    if (issuer) tdm_load_A(gbase, ldsA0, tensor_rows);
#endif

#pragma unroll 1
    for (int c = 0; c < NCHUNK; ++c) {
#if HAVE_TDM
        if (issuer) {
            if (c + 1 < NCHUNK) {
                tdm_load_A(gbase + (c + 1) * KC,
                           (((c + 1) & 1) ? ldsA1 : ldsA0), tensor_rows);
                // chunk c done when only the just-issued transfer is in flight
                __builtin_amdgcn_s_wait_tensorcnt(1);
            } else {
                __builtin_amdgcn_s_wait_tensorcnt(0);
            }
        }
        __syncthreads();                 // publish chunk c to all waves
#else
        // Fallback: cooperative synchronous copy with row clamp
        for (int idx = tid; idx < ROWS_PER_BLOCK * KC; idx += THREADS) {
            int r = idx / KC, k = idx % KC;
            long long row = blockRowBase + r;
            float v = 0.0f;
            if (row < Nrows) v = node_rep[(size_t)row * D_IN + c * KC + k];
            sh_A[c & 1][r * ASTRIDE + k] = v;
        }
        __syncthreads();
#endif
        const float* smA = &sh_A[c & 1][0];
        const int kbase = c * KC;
#pragma unroll
        for (int kb = 0; kb < KC; kb += 4) {
            v2f a = *(const v2f*)(smA + rowOff + kb);
            const int bo = (kbase + kb) * H_;
            v2f bv0, bv1;
            bv0.x = bptr0[bo];        bv0.y = bptr0[bo + H_];
            bv1.x = bptr1[bo];        bv1.y = bptr1[bo + H_];
            c0 = __builtin_amdgcn_wmma_f32_16x16x4_f32(false, a, false, bv0,
                                                       (short)0, c0, false, false);
            c1 = __builtin_amdgcn_wmma_f32_16x16x4_f32(false, a, false, bv1,
                                                       (short)0, c1, false, false);
        }
        __syncthreads();                 // all waves done reading before refill
    }

    // C/D layout: VGPR r, lanes 0-15 -> M=r, N=lane ; lanes 16-31 -> M=r+8, N=lane-16
    {
        const int rbase = wave * 16 + ((lane >> 4) * 8);
        const int col = lane & 15;
#pragma unroll
        for (int r = 0; r < 8; ++r) {
            sh_hidden[(rbase + r) * HSTRIDE + col]      = c0[r];
            sh_hidden[(rbase + r) * HSTRIDE + col + 16] = c1[r];
        }
    }
    __syncthreads();

    // ---- phase 2: KAN layers, one thread per row ----
    if (tid < ROWS_PER_BLOCK) {
        const long long n = blockRowBase + tid;
        float acc[W1];
#pragma unroll
        for (int o = 0; o < W1; ++o) acc[o] = sh_b0[o];

        float Bv[NB];
#pragma unroll 4
        for (int i = 0; i < H_; ++i) {
            float x = sh_hidden[tid * HSTRIDE + i] + sh_mlpb[i];
            float sx = silu(x);
            bspline12(x, Bv);
#pragma unroll
            for (int o = 0; o < W1; ++o) {
                float s = 0.0f;
                const float* cp = &sh_coef0[(i * W1 + o) * NB];
#pragma unroll
                for (int b = 0; b < NB; ++b) s = fmaf(Bv[b], cp[b], s);
                acc[o] = fmaf(sx, sh_wb0[i * W1 + o],
                              fmaf(s, sh_ws0[i * W1 + o], acc[o]));
            }
        }

        float result = sh_b1;
#pragma unroll
        for (int j = 0; j < W1; ++j) {
            float x = acc[j];
            float sx = silu(x);
            bspline12(x, Bv);
            float s = 0.0f;
            const float* cp = &sh_coef1[j * NB];
#pragma unroll
            for (int b = 0; b < NB; ++b) s = fmaf(Bv[b], cp[b], s);
            result = fmaf(sx, sh_wb1[j], fmaf(s, sh_ws1[j], result));
        }

        if (n < Nrows) out[n] = result;
    }
}

extern "C" void kernel_launch(void* const* d_in, const int* in_sizes, int n_in,
                              void* d_out, int out_size, void* d_ws, size_t ws_size,
                              hipStream_t stream) {
    const float* node_rep = (const float*)d_in[0];
    const float* mlp_w    = (const float*)d_in[1];
    const float* mlp_b    = (const float*)d_in[2];
    const float* coef0    = (const float*)d_in[3];
    const float* wb0      = (const float*)d_in[4];
    const float* ws0      = (const float*)d_in[5];
    const float* b0       = (const float*)d_in[6];
    const float* coef1    = (const float*)d_in[7];
    const float* wb1      = (const float*)d_in[8];
    const float* ws1      = (const float*)d_in[9];
    const float* b1       = (const float*)d_in[10];
    float* out = (float*)d_out;

    const int Nrows = in_sizes[0] / D_IN;
    const int blocks = (Nrows + ROWS_PER_BLOCK - 1) / ROWS_PER_BLOCK;

    kan_fused_kernel<<<blocks, THREADS, 0, stream>>>(
        node_rep, mlp_w, mlp_b, coef0, wb0, ws0, b0, coef1, wb1, ws1, b1,
        out, Nrows);
}